// LinearMemristorLayer_8907762172026
// MI455X (gfx1250) — compile-verified
//
#include <hip/hip_runtime.h>
#include <hip/hip_bf16.h>

// y = x @ w + b  (the memristor conductance mapping cancels exactly; see analysis)
// fp32 WMMA path: V_WMMA_F32_16X16X4_F32, one wave per 16x32 output strip.

typedef __attribute__((ext_vector_type(2))) float v2f;
typedef __attribute__((ext_vector_type(8))) float v8f;

#define N_IN  1024
#define N_OUT 512
#define BATCH 128

__global__ __launch_bounds__(32)
void memristor_linear_wmma_f32(const float* __restrict__ X,   // (BATCH, N_IN)
                               const float* __restrict__ W,   // (N_IN, N_OUT)
                               const float* __restrict__ Bv,  // (N_OUT,)
                               float* __restrict__ Y) {       // (BATCH, N_OUT)
  const int lane = threadIdx.x & 31;
  const int h    = lane >> 4;     // lane half: 0 or 1
  const int ml   = lane & 15;
  const int m0   = blockIdx.y << 4;   // 16-row M tile
  const int n0   = blockIdx.x << 5;   // 32-col N strip (two 16-wide WMMA tiles)

  // A fragment source: lane holds A[m0+ml][k0+2h .. k0+2h+1] -> contiguous float2
  const float* arow = X + (size_t)(m0 + ml) * N_IN + 2 * h;
  // B fragment sources: vgpr r holds B[k0 + r + 2h][n0 (+16) + ml]
  const float* b0p = W + n0 + ml;
  const float* b1p = W + n0 + 16 + ml;

  v8f acc0 = {};
  v8f acc1 = {};

#pragma unroll 4
  for (int k0 = 0; k0 < N_IN; k0 += 4) {
    const int ka = k0 + 2 * h;

    v2f a = *(const v2f*)(arow + k0);

    v2f b0, b1;
    b0.x = b0p[(size_t)(ka)     * N_OUT];
    b0.y = b0p[(size_t)(ka + 1) * N_OUT];
    b1.x = b1p[(size_t)(ka)     * N_OUT];
    b1.y = b1p[(size_t)(ka + 1) * N_OUT];

    // 8 args: (neg_a, A, neg_b, B, c_mod, C, reuse_a, reuse_b)
    acc0 = __builtin_amdgcn_wmma_f32_16x16x4_f32(false, a, false, b0,
                                                 (short)0, acc0, false, false);
    acc1 = __builtin_amdgcn_wmma_f32_16x16x4_f32(false, a, false, b1,
                                                 (short)0, acc1, false, false);
  }

  const float bias0 = Bv[n0 + ml];
  const float bias1 = Bv[n0 + 16 + ml];

  // C/D layout: vgpr r, lane L -> row m0 + r + 8*(L>>4), col n0 + (L&15)
#pragma unroll
  for (int r = 0; r < 8; ++r) {
    const int m = m0 + r + 8 * h;
    Y[(size_t)m * N_OUT + n0 + ml]      = acc0[r] + bias0;
    Y[(size_t)m * N_OUT + n0 + 16 + ml] = acc1[r] + bias1;
  }
}

extern "C" void kernel_launch(void* const* d_in, const int* in_sizes, int n_in,
                              void* d_out, int out_size, void* d_ws, size_t ws_size,
                              hipStream_t stream) {
  const float* x = (const float*)d_in[0];   // (128, 1024)
  const float* w = (const float*)d_in[1];   // (1024, 512)
  const float* b = (const float*)d_in[2];   // (512,)
  float* y = (float*)d_out;                 // (128, 512)

  dim3 grid(N_OUT / 32, BATCH / 16);        // (16, 8) = 128 waves
  dim3 block(32);
  memristor_linear_wmma_f32<<<grid, block, 0, stream>>>(x, w, b, y);
}